// YoloLoss_70617852281096
// MI455X (gfx1250) — compile-verified
//
#include <hip/hip_runtime.h>
#include <math.h>

#define BS 32
#define TT 20
#define HH 76
#define WW 76
#define HW (HH*WW)             // 5776
#define CELLS_PER_B (3*HW)     // 17328
#define NCELL (BS*CELLS_PER_B) // 554496 = 2166 * 256 exactly
#define NBLK (NCELL/256)       // 2166
#define NC 80
#define EPSC 1e-7f
#define GT_STRIDE 16

typedef __attribute__((ext_vector_type(2))) float v2f;
typedef __attribute__((ext_vector_type(8))) float v8f;

// ANCHORS / stride (stride = 608/76 = 8)
__device__ __constant__ float c_AW[9] = {1.5f, 2.375f, 5.f, 4.5f, 9.5f, 9.f, 17.75f, 24.f, 57.375f};
__device__ __constant__ float c_AH[9] = {2.f, 4.5f, 3.5f, 9.375f, 6.875f, 18.25f, 13.75f, 30.375f, 50.125f};

__device__ __forceinline__ float sigmoidf(float v) { return 1.f / (1.f + expf(-v)); }

__device__ __forceinline__ float bcef(float p, float t) {
  p = fminf(fmaxf(p, EPSC), 1.f - EPSC);
  return -t * logf(p) - (1.f - t) * logf(1.f - p);
}

// Tensor-core block reduction over a zero-padded 8x16 LDS tile of wave
// partials (rows = waves, cols 0..4 = loss terms, cols 5..15 = zero).
// A = ones(16x4) => D[m][n] = sum_k B[k][n] + C[m][n]; packing column n with
// the 8 wave partials across two accumulating WMMAs gives, in lane n (n<5),
// C[0] = block total of term n. Branch-free: all 32 lanes load real LDS data.
__device__ __forceinline__ float wmma_reduce8x16(const float* wsum, int tid) {
  int g = tid & 15, h = tid >> 4;
  v2f A; A.x = 1.f; A.y = 1.f;
  v8f C = {};
  v2f B;
  B.x = wsum[(h ? 1 : 0) * 16 + g];
  B.y = wsum[(h ? 3 : 2) * 16 + g];
  C = __builtin_amdgcn_wmma_f32_16x16x4_f32(false, A, false, B, (short)0, C, false, false);
  B.x = wsum[(h ? 5 : 4) * 16 + g];
  B.y = wsum[(h ? 7 : 6) * 16 + g];
  C = __builtin_amdgcn_wmma_f32_16x16x4_f32(false, A, false, B, (short)0, C, false, false);
  return C[0];
}

// ---- Kernel 1: per-GT precompute (anchor argmax, cell indices, corners) ----
__global__ void yolo_prep(const float* __restrict__ tgt, const int* __restrict__ lp,
                          float* __restrict__ gtab) {
  int idx = blockIdx.x * blockDim.x + threadIdx.x;
  if (idx >= BS * TT) return;
  int l = lp[0];
  int m0 = 6 - 3 * l;  // ANCHORS_MASK[l][0]
  const float* tr = tgt + idx * 5;
  float gx = tr[0] * WW, gy = tr[1] * HH, gw = tr[2] * WW, gh = tr[3] * HH;
  float best = -1.f; int bn = 0;
  for (int a = 0; a < 9; ++a) {
    float inter = fminf(gw, c_AW[a]) * fminf(gh, c_AH[a]);
    float uni = gw * gh + c_AW[a] * c_AH[a] - inter;
    float r = inter / uni;
    if (r > best) { best = r; bn = a; }   // first max, like jnp.argmax
  }
  int k = bn - m0;
  float valid = (k >= 0 && k < 3) ? 1.f : 0.f;
  int gi = (int)floorf(gx); gi = gi < 0 ? 0 : (gi > WW - 1 ? WW - 1 : gi);
  int gj = (int)floorf(gy); gj = gj < 0 ? 0 : (gj > HH - 1 ? HH - 1 : gj);
  float* o = gtab + idx * GT_STRIDE;
  o[0] = gx; o[1] = gy; o[2] = gw; o[3] = gh;
  o[4] = (float)gi; o[5] = (float)gj; o[6] = (float)k; o[7] = valid; o[8] = tr[4];
  o[9]  = gx - gw * 0.5f;   // g0x
  o[10] = gx + gw * 0.5f;   // g1x
  o[11] = gy - gh * 0.5f;   // g0y
  o[12] = gy + gh * 0.5f;   // g1y
  o[13] = gw * gh;          // area
  o[14] = 0.f; o[15] = 0.f;
}

// ---- Kernel 2: per-cell loss terms + WMMA block reduction ----
__global__ void __launch_bounds__(256) yolo_main(const float* __restrict__ inp,
                                                 const int* __restrict__ lp,
                                                 const float* __restrict__ gtab,
                                                 float* __restrict__ part) {
  __shared__ float sgt[2 * TT * GT_STRIDE];  // up to 2 batches of GT records
  __shared__ float wsum[8 * 16];             // zero-padded wave partials
  int tid = threadIdx.x;
  int idx = blockIdx.x * 256 + tid;          // exact cover, no early return
  int b0 = (blockIdx.x * 256) / CELLS_PER_B;

  int nb2 = (BS - b0) < 2 ? (BS - b0) : 2;
  for (int p = tid; p < nb2 * TT * GT_STRIDE; p += 256)
    sgt[p] = gtab[b0 * TT * GT_STRIDE + p];
  if (tid < 8 * 16) wsum[tid] = 0.f;
  __syncthreads();

  int i = idx % WW;
  int j = (idx / WW) % HH;
  int k = (idx / HW) % 3;
  int b = idx / CELLS_PER_B;
  int l = lp[0];
  int m0 = 6 - 3 * l;

  int cbase = b * 255 + k * 85;
  int sp = j * WW + i;
  float rx = inp[(cbase + 0) * HW + sp];
  float ry = inp[(cbase + 1) * HW + sp];
  float rw = inp[(cbase + 2) * HW + sp];
  float rh = inp[(cbase + 3) * HW + sp];
  float rc = inp[(cbase + 4) * HW + sp];
  float sx = sigmoidf(rx), sy = sigmoidf(ry), conf = sigmoidf(rc);
  float aw = c_AW[m0 + k], ah = c_AH[m0 + k];
  float px = (float)i + sx, py = (float)j + sy;
  float pw = expf(rw) * aw, ph = expf(rh) * ah;
  float p0x = px - pw * 0.5f, p1x = px + pw * 0.5f;
  float p0y = py - ph * 0.5f, p1y = py + ph * 0.5f;
  float pa = pw * ph;

  const float* g = sgt + (b - b0) * TT * GT_STRIDE;
  float bestiou = 0.f;
  int is_obj = 0, tsel = 0;
  #pragma unroll 4
  for (int t = 0; t < TT; ++t) {
    const float* r = g + t * GT_STRIDE;
    // hot loop: precomputed GT corners + area (11 VALU ops/iter)
    float iw = fmaxf(fminf(r[10], p1x) - fmaxf(r[9],  p0x), 0.f);
    float ih = fmaxf(fminf(r[12], p1y) - fmaxf(r[11], p0y), 0.f);
    float inter = iw * ih;
    float iou = inter / (r[13] + pa - inter);          // reference: no clip here
    bestiou = fmaxf(bestiou, iou);
    if (r[7] > 0.5f && (int)r[6] == k && (int)r[4] == i && (int)r[5] == j) {
      is_obj = 1; tsel = t;                            // last match wins (scatter-set)
    }
  }

  float v0 = 0.f, v2 = 0.f;
  if (is_obj) {
    const float* r = g + tsel * GT_STRIDE;
    float tx = r[0], ty = r[1], tw = r[2], th = r[3];
    float t0x = r[9], t1x = r[10], t0y = r[11], t1y = r[12], ta = r[13];
    float iw = fmaxf(fminf(t1x, p1x) - fmaxf(t0x, p0x), 0.f);
    float ih = fmaxf(fminf(t1y, p1y) - fmaxf(t0y, p0y), 0.f);
    float inter = iw * ih;
    float iou = inter / fmaxf(pa + ta - inter, 1e-6f);
    float cd = (px - tx) * (px - tx) + (py - ty) * (py - ty);
    float ew = fmaxf(fmaxf(t1x, p1x) - fminf(t0x, p0x), 0.f);
    float eh = fmaxf(fmaxf(t1y, p1y) - fminf(t0y, p0y), 0.f);
    float ed = ew * ew + eh * eh;
    float ciou = iou - cd / fmaxf(ed, 1e-6f);
    float dv = atanf(pw / fmaxf(ph, 1e-6f)) - atanf(tw / fmaxf(th, 1e-6f));
    float vv = 0.40528473456935109f * dv * dv;         // 4/pi^2
    float alpha = vv / fmaxf(1.f - iou + vv, 1e-6f);
    ciou -= alpha * vv;
    v0 = 1.f - ciou;
    // class BCE: only object cells contribute -> skip 80/85 of input bandwidth
    int tcls = (int)r[8];
    float s = 0.f;
    int cb = (cbase + 5) * HW + sp;
    for (int c = 0; c < NC; ++c) {
      float p = sigmoidf(inp[cb + c * HW]);
      s += bcef(p, (c == tcls) ? 1.f : 0.f);
    }
    v2 = s;
  }
  int noobj = (!is_obj) && (bestiou <= 0.5f);          // ignore-thresh suppression
  int mask = is_obj || noobj;
  float v1 = is_obj ? 1.f : 0.f;
  float v3 = mask ? bcef(conf, v1) : 0.f;
  float v4 = mask ? 1.f : 0.f;

  float v[5] = {v0, v1, v2, v3, v4};
  #pragma unroll
  for (int off = 16; off > 0; off >>= 1)
    #pragma unroll
    for (int a = 0; a < 5; ++a) v[a] += __shfl_xor(v[a], off, 32);
  int lane = tid & 31, wid = tid >> 5;
  if (lane == 0)
    for (int a = 0; a < 5; ++a) wsum[wid * 16 + a] = v[a];
  __syncthreads();
  if (tid < 32) {                                      // full wave -> EXEC all-ones
    float tot = wmma_reduce8x16(wsum, tid);
    if (tid < 5) part[blockIdx.x * 5 + tid] = tot;
  }
}

// ---- Kernel 3: global reduction + final scalar ----
__global__ void __launch_bounds__(256) yolo_final(const float* __restrict__ part,
                                                  const int* __restrict__ lp,
                                                  float* __restrict__ out) {
  __shared__ float wsum[8 * 16];
  __shared__ float tot[5];
  int tid = threadIdx.x;
  if (tid < 8 * 16) wsum[tid] = 0.f;
  float v[5] = {0.f, 0.f, 0.f, 0.f, 0.f};
  for (int p = tid; p < NBLK; p += 256)
    for (int a = 0; a < 5; ++a) v[a] += part[p * 5 + a];
  #pragma unroll
  for (int off = 16; off > 0; off >>= 1)
    #pragma unroll
    for (int a = 0; a < 5; ++a) v[a] += __shfl_xor(v[a], off, 32);
  int lane = tid & 31, wid = tid >> 5;
  __syncthreads();   // zero-fill complete before leader writes
  if (lane == 0)
    for (int a = 0; a < 5; ++a) wsum[wid * 16 + a] = v[a];
  __syncthreads();
  if (tid < 32) {
    float t = wmma_reduce8x16(wsum, tid);
    if (tid < 5) tot[tid] = t;
  }
  __syncthreads();
  if (tid == 0) {
    float nobj = fmaxf(tot[1], 1.f);
    float msk  = fmaxf(tot[4], 1.f);
    int l = lp[0];
    float bal = (l == 0) ? 0.4f : ((l == 1) ? 1.f : 4.f);
    const float OBJR = 5.f * (608.f * 608.f) / (416.f * 416.f);
    const float CLSR = 1.f;                            // 80/80
    out[0] = tot[0] / nobj * 0.05f
           + tot[2] / (nobj * 80.f) * CLSR
           + tot[3] / msk * bal * OBJR;
  }
}

extern "C" void kernel_launch(void* const* d_in, const int* in_sizes, int n_in,
                              void* d_out, int out_size, void* d_ws, size_t ws_size,
                              hipStream_t stream) {
  const float* inp = (const float*)d_in[0];
  const float* tgt = (const float*)d_in[1];
  const int*   lp  = (const int*)d_in[2];
  float* ws   = (float*)d_ws;
  float* gtab = ws;                          // 640 * 16 floats
  float* part = ws + BS * TT * GT_STRIDE;    // NBLK * 5 floats
  yolo_prep<<<(BS * TT + 255) / 256, 256, 0, stream>>>(tgt, lp, gtab);
  yolo_main<<<NBLK, 256, 0, stream>>>(inp, lp, gtab, part);
  yolo_final<<<1, 256, 0, stream>>>(part, lp, (float*)d_out);
}